// RIRSynthesiser_60378650247708
// MI455X (gfx1250) — compile-verified
//
#include <hip/hip_runtime.h>
#include <math.h>
#include <stdint.h>

// ---------------------------------------------------------------------------
// RIR synthesiser for MI455X (gfx1250, wave32, WMMA + TDM + async-LDS).
// Shapes: B=64, FEAT=16, T=32001, BANDS=8, ND=16, OUT_LEN=32000, K=1024(+43)
// ---------------------------------------------------------------------------

typedef __attribute__((ext_vector_type(2))) float v2f;
typedef __attribute__((ext_vector_type(8))) float v8f;
typedef __attribute__((ext_vector_type(4))) unsigned int v4u;
typedef __attribute__((ext_vector_type(4))) int v4i;
typedef __attribute__((ext_vector_type(8))) int v8i;

#define NB        64
#define TN        256008      // T * BANDS
#define TT        32001
#define TTP       32016       // padded edc_1d row stride (16-float aligned rows)
#define OUT_LEN   32000
#define NBLK      2001        // ceil(32001/16) signal blocks
#define OBLK      2000        // 32000/16 output blocks

__device__ __forceinline__ v8f wmma_f32_4(v2f a, v2f b, v8f c) {
  // D = A(16x4 f32) x B(4x16 f32) + C(16x16 f32)
  return __builtin_amdgcn_wmma_f32_16x16x4_f32(
      /*neg_a=*/false, a, /*neg_b=*/false, b,
      /*c_mod=*/(short)0, c, /*reuse_a=*/false, /*reuse_b=*/false);
}

// Async global->LDS copy, 16 bytes per lane, tracked by ASYNCcnt.
__device__ __forceinline__ void async_load_b128(void* lds_dst, const void* gsrc) {
  unsigned lds = (unsigned)(uintptr_t)lds_dst;   // low 32 bits = LDS offset
  asm volatile("global_load_async_to_lds_b128 %0, %1, off"
               :: "v"(lds), "v"(gsrc) : "memory");
}
__device__ __forceinline__ void wait_asynccnt0() {
  asm volatile("s_wait_asynccnt 0x0" ::: "memory");
}

// TDM: 1D tensor (nelem f32) global -> LDS.  D# per CDNA5 ISA 8.3/8.4.
__device__ __forceinline__ void tdm_load_1d_f32(unsigned lds_off,
                                                const void* gaddr,
                                                unsigned nelem) {
  unsigned long long ga = (unsigned long long)(uintptr_t)gaddr;
  v4u g0;
  g0.x = 1u;                                        // count=1, no gather
  g0.y = lds_off;                                   // lds_addr  [63:32]
  g0.z = (unsigned)(ga & 0xffffffffu);              // global_addr[31:0]
  g0.w = (unsigned)((ga >> 32) & 0x01ffffffu)       // global_addr[56:32]
         | 0x80000000u;                             // type=2 ("image")
  v8i g1;
  g1[0] = (int)(2u << 16);                          // data_size=2 (4 bytes)
  g1[1] = (int)((nelem & 0xffffu) << 16);           // tensor_dim0 lo16 @63:48
  g1[2] = (int)(((nelem >> 16) & 0xffffu) | (1u << 16)); // dim0 hi | tensor_dim1=1
  g1[3] = (int)((nelem & 0xffffu) << 16);           // tile_dim0 @127:112
  g1[4] = 1;                                        // tile_dim1=1
  g1[5] = (int)nelem;                               // tensor_dim0_stride lo32
  g1[6] = 0;
  g1[7] = 0;
  v4i z4 = {0, 0, 0, 0};
#if defined(__clang_major__) && (__clang_major__ >= 23)
  v8i z8 = {0, 0, 0, 0, 0, 0, 0, 0};
  __builtin_amdgcn_tensor_load_to_lds(g0, g1, z4, z4, z8, 0);
#else
  __builtin_amdgcn_tensor_load_to_lds(g0, g1, z4, z4, 0);
#endif
}

// ---------------------------------------------------------------------------
// K1: h1 = relu(x @ lstm_W1 + b1)   (64,16)x(16,64) -> (64,64)
// ---------------------------------------------------------------------------
__global__ __launch_bounds__(256) void k1_front(const float* __restrict__ x,
                                                const float* __restrict__ W1,
                                                const float* __restrict__ b1,
                                                float* __restrict__ h1) {
  int i = blockIdx.x * 256 + threadIdx.x;          // 0..4095
  int bi = i >> 6, o = i & 63;
  float acc = b1[o];
#pragma unroll
  for (int k = 0; k < 16; ++k) acc = fmaf(x[bi * 16 + k], W1[k * 64 + o], acc);
  h1[i] = fmaxf(acc, 0.f);
}

// ---------------------------------------------------------------------------
// K2: edc_pred = h1 @ lstm_W2 + b2   (64,64)x(64,256008)
// One wave per 16-wide N tile; all four M tiles (batch 0..63) per wave so W2
// is streamed exactly once.  K=64 via 16 chained V_WMMA_F32_16X16X4_F32.
// ---------------------------------------------------------------------------
__global__ __launch_bounds__(32) void k2_gemm(const float* __restrict__ h1,
                                              const float* __restrict__ W2,
                                              const float* __restrict__ b2,
                                              float* __restrict__ edc_pred) {
  const int n0   = blockIdx.x * 16;
  const int lane = threadIdx.x;
  const int col  = lane & 15;
  const int half = lane >> 4;
  const int n    = n0 + col;
  const int nc   = (n < TN) ? n : (TN - 1);        // clamped for loads

  v8f c0 = {}, c1 = {}, c2 = {}, c3 = {};
#pragma unroll
  for (int k0 = 0; k0 < 64; k0 += 4) {
    const int kb = k0 + 2 * half;                  // A:K0/K1 vs K2/K3 split
    // B tile: B[kk][col] = W2[kb+kk][n]
    v2f b;
    b.x = W2[(size_t)kb * TN + nc];
    b.y = W2[(size_t)(kb + 1) * TN + nc];
    if (k0 < 60) {                                 // stream-ahead of next K rows
      __builtin_prefetch(&W2[(size_t)(kb + 4) * TN + nc], 0, 1);
      __builtin_prefetch(&W2[(size_t)(kb + 5) * TN + nc], 0, 1);
    }
    // A tiles for the 4 batch-row groups
    v2f a0, a1, a2, a3;
    const float* r0 = h1 + (col)      * 64 + kb;
    const float* r1 = h1 + (16 + col) * 64 + kb;
    const float* r2 = h1 + (32 + col) * 64 + kb;
    const float* r3 = h1 + (48 + col) * 64 + kb;
    a0.x = r0[0]; a0.y = r0[1];
    a1.x = r1[0]; a1.y = r1[1];
    a2.x = r2[0]; a2.y = r2[1];
    a3.x = r3[0]; a3.y = r3[1];
    c0 = wmma_f32_4(a0, b, c0);
    c1 = wmma_f32_4(a1, b, c1);
    c2 = wmma_f32_4(a2, b, c2);
    c3 = wmma_f32_4(a3, b, c3);
  }
  if (n < TN) {
    const float bias = b2[n];
#pragma unroll
    for (int e = 0; e < 8; ++e) {
      const int row = e + 8 * half;                // C/D layout: half selects M 0-7 / 8-15
      edc_pred[(size_t)(row)      * TN + n] = c0[e] + bias;
      edc_pred[(size_t)(16 + row) * TN + n] = c1[e] + bias;
      edc_pred[(size_t)(32 + row) * TN + n] = c2[e] + bias;
      edc_pred[(size_t)(48 + row) * TN + n] = c3[e] + bias;
    }
  }
}

// ---------------------------------------------------------------------------
// K3a: edc_1d[b][t] = mean over 8 bands  (row stride padded to TTP)
// ---------------------------------------------------------------------------
__global__ __launch_bounds__(256) void k3a_bandmean(const float* __restrict__ edc_pred,
                                                    float* __restrict__ edc_1d) {
  size_t i = (size_t)blockIdx.x * 256 + threadIdx.x;
  if (i >= (size_t)NB * TT) return;
  int bi = (int)(i / TT), t = (int)(i % TT);
  const float* p = edc_pred + (size_t)bi * TN + (size_t)t * 8;
  float s = 0.f;
#pragma unroll
  for (int k = 0; k < 8; ++k) s += p[k];
  edc_1d[(size_t)bi * TTP + t] = s * 0.125f;
}

// ---------------------------------------------------------------------------
// K3b: edc_feat[b] = [mean_t(edc_pred), q4 - q1]   -> (64,16)
// ---------------------------------------------------------------------------
__global__ __launch_bounds__(256) void k3b_feat(const float* __restrict__ edc_pred,
                                                float* __restrict__ feat) {
  __shared__ float r0[256], r1[256], r4[256];
  const int bi = blockIdx.x >> 3, band = blockIdx.x & 7;
  const int tid = threadIdx.x;
  float s = 0.f, sq1 = 0.f, sq4 = 0.f;
  for (int t = tid; t < TT; t += 256) {
    float v = edc_pred[(size_t)bi * TN + (size_t)t * 8 + band];
    s += v;
    if (t < 8000) sq1 += v;          // T//4 = 8000
    if (t >= 24000) sq4 += v;        // 3*T//4 = 24000 -> 8001 samples
  }
  r0[tid] = s; r1[tid] = sq1; r4[tid] = sq4;
  __syncthreads();
  for (int off = 128; off > 0; off >>= 1) {
    if (tid < off) { r0[tid] += r0[tid + off]; r1[tid] += r1[tid + off]; r4[tid] += r4[tid + off]; }
    __syncthreads();
  }
  if (tid == 0) {
    feat[bi * 16 + band]     = r0[0] / 32001.f;
    feat[bi * 16 + 8 + band] = r4[0] / 8001.f - r1[0] / 8000.f;
  }
}

// ---------------------------------------------------------------------------
// K4: mapper MLP chain (all tiny) -> log_kappa, alpha_raw, beta_raw
// ---------------------------------------------------------------------------
__global__ __launch_bounds__(256) void k4_mapper(
    const float* __restrict__ x, const float* __restrict__ feat,
    const float* __restrict__ eW1, const float* __restrict__ eb1,
    const float* __restrict__ eW2, const float* __restrict__ eb2,
    const float* __restrict__ dW1, const float* __restrict__ db1,
    const float* __restrict__ dW2, const float* __restrict__ db2,
    const float* __restrict__ aW1, const float* __restrict__ ab1,
    const float* __restrict__ aW2, const float* __restrict__ ab2,
    const float* __restrict__ bW1, const float* __restrict__ bb1,
    const float* __restrict__ bW2, const float* __restrict__ bb2,
    float* __restrict__ log_kappa, float* __restrict__ alpha_raw,
    float* __restrict__ beta_raw) {
  extern __shared__ float sm[];
  float* s_feat = sm;            // 1024
  float* s_t1   = sm + 1024;     // 4096
  float* s_h    = sm + 5120;     // 4096
  float* s_din  = sm + 9216;     // 64*68 = 4352 (padded 67->68)
  float* s_t2   = sm + 13568;    // 4096
  float* s_t3   = sm + 17664;    // 2048
  const int tid = threadIdx.x;

  for (int i = tid; i < 1024; i += 256) s_feat[i] = feat[i];
  __syncthreads();
  for (int i = tid; i < 4096; i += 256) {          // t1 = relu(feat@eW1+eb1)
    int bi = i >> 6, o = i & 63;
    float acc = eb1[o];
#pragma unroll
    for (int k = 0; k < 16; ++k) acc = fmaf(s_feat[bi * 16 + k], eW1[k * 64 + o], acc);
    s_t1[i] = fmaxf(acc, 0.f);
  }
  __syncthreads();
  for (int i = tid; i < 4096; i += 256) {          // h = relu(t1@eW2+eb2)
    int bi = i >> 6, o = i & 63;
    float acc = eb2[o];
    for (int k = 0; k < 64; ++k) acc = fmaf(s_t1[bi * 64 + k], eW2[k * 64 + o], acc);
    s_h[i] = fmaxf(acc, 0.f);
  }
  __syncthreads();
  for (int i = tid; i < 64 * 67; i += 256) {       // din = [x[:, :3], h]
    int bi = i / 67, j = i % 67;
    s_din[bi * 68 + j] = (j < 3) ? x[bi * 16 + j] : s_h[bi * 64 + (j - 3)];
  }
  __syncthreads();
  for (int i = tid; i < 4096; i += 256) {          // t2 = relu(din@dW1+db1)
    int bi = i >> 6, o = i & 63;
    float acc = db1[o];
    for (int k = 0; k < 67; ++k) acc = fmaf(s_din[bi * 68 + k], dW1[k * 64 + o], acc);
    s_t2[i] = fmaxf(acc, 0.f);
  }
  __syncthreads();
  for (int i = tid; i < 1024; i += 256) {          // log_kappa = t2@dW2+db2
    int bi = i >> 4, o = i & 15;
    float acc = db2[o];
    for (int k = 0; k < 64; ++k) acc = fmaf(s_t2[bi * 64 + k], dW2[k * 16 + o], acc);
    log_kappa[i] = acc;
  }
  for (int i = tid; i < 2048; i += 256) {          // t3 = relu(h@aW1+ab1)
    int bi = i >> 5, o = i & 31;
    float acc = ab1[o];
    for (int k = 0; k < 64; ++k) acc = fmaf(s_h[bi * 64 + k], aW1[k * 32 + o], acc);
    s_t3[i] = fmaxf(acc, 0.f);
  }
  __syncthreads();
  for (int i = tid; i < 1024; i += 256) {          // alpha_raw
    int bi = i >> 4, o = i & 15;
    float acc = ab2[o];
    for (int k = 0; k < 32; ++k) acc = fmaf(s_t3[bi * 32 + k], aW2[k * 16 + o], acc);
    alpha_raw[i] = acc;
  }
  __syncthreads();
  for (int i = tid; i < 2048; i += 256) {          // t3 = relu(h@bW1+bb1)
    int bi = i >> 5, o = i & 31;
    float acc = bb1[o];
    for (int k = 0; k < 64; ++k) acc = fmaf(s_h[bi * 64 + k], bW1[k * 32 + o], acc);
    s_t3[i] = fmaxf(acc, 0.f);
  }
  __syncthreads();
  for (int i = tid; i < 1024; i += 256) {          // beta_raw
    int bi = i >> 4, o = i & 15;
    float acc = bb2[o];
    for (int k = 0; k < 32; ++k) acc = fmaf(s_t3[bi * 32 + k], bW2[k * 16 + o], acc);
    beta_raw[i] = acc;
  }
}

// ---------------------------------------------------------------------------
// K5: sticky-sign parity.  sign[b][t] = 1-2*((cumsum flips, flips[:,0]=0) & 1)
// ---------------------------------------------------------------------------
__global__ __launch_bounds__(1024) void k5_sign(const int* __restrict__ flips,
                                                float* __restrict__ sgn) {
  __shared__ int s[1024];
  __shared__ int carry_s;
  const int b = blockIdx.x, tid = threadIdx.x;
  if (tid == 0) carry_s = 0;
  __syncthreads();
  for (int base = 0; base < OUT_LEN; base += 1024) {
    const int t = base + tid;
    s[tid] = (t > 0 && t < OUT_LEN) ? (flips[(size_t)b * OUT_LEN + t] & 1) : 0;
    __syncthreads();
    for (int off = 1; off < 1024; off <<= 1) {     // Hillis-Steele XOR scan
      int tmp = (tid >= off) ? s[tid - off] : 0;
      __syncthreads();
      s[tid] ^= tmp;
      __syncthreads();
    }
    const int par = s[tid] ^ carry_s;
    if (t < OUT_LEN) sgn[(size_t)b * OUT_LEN + t] = 1.f - 2.f * (float)par;
    __syncthreads();
    if (tid == 1023) carry_s ^= s[1023];
    __syncthreads();
  }
}

// ---------------------------------------------------------------------------
// K6: fused causal conv (1024 taps = late + early) + phase -> rir.
// One workgroup per batch row.  LDS staging uses CDNA5 data movers:
//   - taps via TDM tensor_load_to_lds (TENSORcnt)
//   - signal body via global_load_async_to_lds_b128 (ASYNCcnt)
// Block-Toeplitz WMMA: D_{v,g}[j][c] = sum_r sig[16v+j-r]*taps[256g+16c+r]
// accumulates into out block u = v + c + 16g; rounds over v mod 64 keep
// cross-wave output windows disjoint -> no LDS atomics.
// ---------------------------------------------------------------------------
__global__ __launch_bounds__(256) void k6_conv(const float* __restrict__ edc_1d,
                                               const float* __restrict__ sgn,
                                               const float* __restrict__ late_kernel,
                                               const float* __restrict__ early_gains,
                                               float* __restrict__ rir) {
  extern __shared__ float smem[];
  float* taps = smem;                    // 1024
  float* sig  = smem + 1024 + 16;        // valid for index in [-16, 32016)
  float* out  = smem + 1024 + 32032;     // 32000
  const int b = blockIdx.x, tid = threadIdx.x;
  const int wave = tid >> 5, lane = tid & 31;
  const int col = lane & 15, half = lane >> 4;

  // --- TDM: taps <- late_kernel (1024 f32), one wave issues ---
  if (wave == 0) tdm_load_1d_f32((unsigned)(uintptr_t)taps, late_kernel, 1024u);

  // --- async copies: sig[0..31999] <- edc_1d row (8000 x b128, aligned) ---
  const float* src = edc_1d + (size_t)b * TTP;
  for (int q = tid; q < 8000; q += 256)
    async_load_b128(&sig[4 * q], src + 4 * q);

  // --- pads, straggler, output zeroing (regular path) ---
  if (tid < 16) sig[(int)tid - 16] = 0.f;          // front pad
  if (tid < 15) sig[TT + tid] = 0.f;               // tail pad 32001..32015
  if (tid == 255) sig[32000] = src[32000];
  for (int i = tid; i < OUT_LEN; i += 256) out[i] = 0.f;

  if (wave == 0) __builtin_amdgcn_s_wait_tensorcnt(0);
  wait_asynccnt0();
  __syncthreads();

  if (tid < 43) taps[tid] += early_gains[tid];     // fold early reflections
  __syncthreads();

  // Hoist B tiles (taps are v-invariant): B[kk][c] = taps[256g+16c+kb+kk]
  v2f tb0[4], tb1[4], tb2[4], tb3[4];
#pragma unroll
  for (int kk = 0; kk < 4; ++kk) {
    const int y = 16 * col + 4 * kk + 2 * half;
    tb0[kk].x = taps[y];       tb0[kk].y = taps[y + 1];
    tb1[kk].x = taps[256 + y]; tb1[kk].y = taps[256 + y + 1];
    tb2[kk].x = taps[512 + y]; tb2[kk].y = taps[512 + y + 1];
    tb3[kk].x = taps[768 + y]; tb3[kk].y = taps[768 + y + 1];
  }

  // A VGPR layout: lane<16 -> {K0,K1} of row M=col; lane>=16 -> {K2,K3}
  const int abase = col - 2 * half;
  for (int v0 = 0; v0 < 64; ++v0) {
    for (int v = v0 + (wave << 6); v < NBLK; v += 512) {
      const int sb = 16 * v + abase;
      v8f d0 = {}, d1 = {}, d2 = {}, d3 = {};
#pragma unroll
      for (int kk = 0; kk < 4; ++kk) {
        v2f a;
        const int xx = sb - 4 * kk;
        a.x = sig[xx];
        a.y = sig[xx - 1];
        d0 = wmma_f32_4(a, tb0[kk], d0);
        d1 = wmma_f32_4(a, tb1[kk], d1);
        d2 = wmma_f32_4(a, tb2[kk], d2);
        d3 = wmma_f32_4(a, tb3[kk], d3);
      }
      // scatter: lane holds column c=col -> output block u = v + col + 16g,
      // element e -> j = e + 8*half.  Per-lane addresses distinct; rounds make
      // cross-wave windows disjoint, so plain read-modify-write is race-free.
      int u = v + col;
      if (u < OBLK) {
#pragma unroll
        for (int e = 0; e < 8; ++e) out[16 * u + e + 8 * half] += d0[e];
      }
      u += 16;
      if (u < OBLK) {
#pragma unroll
        for (int e = 0; e < 8; ++e) out[16 * u + e + 8 * half] += d1[e];
      }
      u += 16;
      if (u < OBLK) {
#pragma unroll
        for (int e = 0; e < 8; ++e) out[16 * u + e + 8 * half] += d2[e];
      }
      u += 16;
      if (u < OBLK) {
#pragma unroll
        for (int e = 0; e < 8; ++e) out[16 * u + e + 8 * half] += d3[e];
      }
    }
    __syncthreads();
  }

  for (int t = tid; t < OUT_LEN; t += 256) {
    const float diff = sig[t] - sig[t + 1];
    const float amp = diff > 0.f ? sqrtf(diff) : 0.f;
    rir[(size_t)b * OUT_LEN + t] = out[t] * amp * sgn[(size_t)b * OUT_LEN + t];
  }
}

// ---------------------------------------------------------------------------
extern "C" void kernel_launch(void* const* d_in, const int* in_sizes, int n_in,
                              void* d_out, int out_size, void* d_ws, size_t ws_size,
                              hipStream_t stream) {
  const float* x        = (const float*)d_in[0];
  const float* lstm_W1  = (const float*)d_in[1];
  const float* lstm_b1  = (const float*)d_in[2];
  const float* lstm_W2  = (const float*)d_in[3];
  const float* lstm_b2  = (const float*)d_in[4];
  const float* enc_W1   = (const float*)d_in[5];
  const float* enc_b1   = (const float*)d_in[6];
  const float* enc_W2   = (const float*)d_in[7];
  const float* enc_b2   = (const float*)d_in[8];
  const float* del_W1   = (const float*)d_in[9];
  const float* del_b1   = (const float*)d_in[10];
  const float* del_W2   = (const float*)d_in[11];
  const float* del_b2   = (const float*)d_in[12];
  const float* alp_W1   = (const float*)d_in[13];
  const float* alp_b1   = (const float*)d_in[14];
  const float* alp_W2   = (const float*)d_in[15];
  const float* alp_b2   = (const float*)d_in[16];
  const float* bet_W1   = (const float*)d_in[17];
  const float* bet_b1   = (const float*)d_in[18];
  const float* bet_W2   = (const float*)d_in[19];
  const float* bet_b2   = (const float*)d_in[20];
  const float* early_g  = (const float*)d_in[21];
  const float* late_k   = (const float*)d_in[22];
  const int*   flips    = (const int*)d_in[23];

  float* out       = (float*)d_out;
  float* rir       = out;                                 // 64*32000
  float* edc_pred  = out + (size_t)NB * OUT_LEN;          // 64*256008
  float* log_kappa = edc_pred + (size_t)NB * TN;          // 1024
  float* alpha_raw = log_kappa + 1024;
  float* beta_raw  = alpha_raw + 1024;

  float* ws     = (float*)d_ws;
  float* h1     = ws;                            // 4096
  float* feat   = ws + 4096;                     // 1024
  float* edc_1d = ws + 5120;                     // 64*32016 (padded rows)
  float* sgnbuf = edc_1d + (size_t)NB * TTP;     // 64*32000

  k1_front<<<16, 256, 0, stream>>>(x, lstm_W1, lstm_b1, h1);
  k2_gemm<<<(TN + 15) / 16, 32, 0, stream>>>(h1, lstm_W2, lstm_b2, edc_pred);
  {
    const size_t nel = (size_t)NB * TT;
    k3a_bandmean<<<(unsigned)((nel + 255) / 256), 256, 0, stream>>>(edc_pred, edc_1d);
  }
  k3b_feat<<<NB * 8, 256, 0, stream>>>(edc_pred, feat);
  k4_mapper<<<1, 256, 19712 * sizeof(float), stream>>>(
      x, feat, enc_W1, enc_b1, enc_W2, enc_b2, del_W1, del_b1, del_W2, del_b2,
      alp_W1, alp_b1, alp_W2, alp_b2, bet_W1, bet_b1, bet_W2, bet_b2,
      log_kappa, alpha_raw, beta_raw);
  k5_sign<<<NB, 1024, 0, stream>>>(flips, sgnbuf);
  k6_conv<<<NB, 256, (1024 + 32032 + OUT_LEN) * sizeof(float), stream>>>(
      edc_1d, sgnbuf, late_k, early_g, rir);
}